// FFTLMHead_30502857736253
// MI455X (gfx1250) — compile-verified
//
#include <hip/hip_runtime.h>

// FFT LM head collapses algebraically:
//   sum_n ifft(X)[n] == X[k=0]   (DC extraction)
// => logits[t,v] = (sum_j ((h@U)*S)[t,j]) * (sum_j Vh[j,v])  -- a rank-1 outer product.
// Kernel 1: a[t] via WMMA f32 16x16x4 GEMM (h@U) + S-scale + row reduction.
// Kernel 2: b[v] column-sum of Vh + outer product write (bandwidth bound, ~19 MB).

#define D_MODEL 2048
#define RANK    64
#define VOCAB   50257
#define N_TOK   32

typedef __attribute__((ext_vector_type(2))) float v2f;
typedef __attribute__((ext_vector_type(8))) float v8f;

// One workgroup, 8 waves. Wave w computes 16x16 tile (mtile = w>>2, ntile = w&3)
// of C = h @ U, accumulating over K=2048 in steps of 4 with
// V_WMMA_F32_16X16X4_F32. Then scales by S and reduces rows to a[32].
__global__ __launch_bounds__(256) void fftlm_rowsum_wmma(
    const float* __restrict__ h, const float* __restrict__ U,
    const float* __restrict__ S, float* __restrict__ a_out) {
  __shared__ float tile[N_TOK][RANK + 1];  // +1 pad: conflict-free row reduce

  const int tid  = threadIdx.x;
  const int wave = tid >> 5;
  const int lane = tid & 31;
  const int m0 = (wave >> 2) * 16;   // 0 or 16
  const int n0 = (wave & 3) * 16;    // 0,16,32,48

  // A-frag (16x4 f32): lanes 0-15 -> M=lane, K pair {0,1}; lanes 16-31 -> K pair {2,3}
  const int M       = lane & 15;
  const int halfSel = lane >> 4;
  const int kOff    = halfSel * 2;
  // B-frag (4x16 f32): N = lane&15, same K-half split as A.
  const int N = lane & 15;

  const float* __restrict__ hRow = h + (size_t)(m0 + M) * D_MODEL;

  v8f c = {};
  for (int k0 = 0; k0 < D_MODEL; k0 += 4) {
    v2f aF, bF;
    aF.x = hRow[k0 + kOff];
    aF.y = hRow[k0 + kOff + 1];
    bF.x = U[(size_t)(k0 + kOff)     * RANK + n0 + N];
    bF.y = U[(size_t)(k0 + kOff + 1) * RANK + n0 + N];
    // 8 args: (neg_a, A, neg_b, B, c_mod, C, reuse_a, reuse_b)
    c = __builtin_amdgcn_wmma_f32_16x16x4_f32(false, aF, false, bF,
                                              (short)0, c, false, false);
  }

  // C/D layout: VGPR r holds M = r (lanes 0-15) or M = r+8 (lanes 16-31), N = lane&15.
  const float sN    = S[n0 + N];
  const int   mBase = m0 + (halfSel ? 8 : 0);
#pragma unroll
  for (int r = 0; r < 8; ++r)
    tile[mBase + r][n0 + N] = c[r] * sN;

  __syncthreads();

  if (tid < N_TOK) {
    float s = 0.f;
#pragma unroll
    for (int j = 0; j < RANK; ++j) s += tile[tid][j];
    a_out[tid] = s;
  }
}

// Bandwidth-bound phase: b[v] = sum_j Vh[j,v]; out[t,v] = a[t] * b[v].
// One vocab column per thread; all Vh reads and out writes are unit-stride
// across the wave (coalesced b32).
__global__ __launch_bounds__(256) void fftlm_outer(
    const float* __restrict__ Vh, const float* __restrict__ a,
    float* __restrict__ out) {
  __shared__ float aSh[N_TOK];
  if (threadIdx.x < N_TOK) aSh[threadIdx.x] = a[threadIdx.x];
  __syncthreads();

  const int v = blockIdx.x * blockDim.x + threadIdx.x;
  if (v >= VOCAB) return;

  float b = 0.f;
#pragma unroll
  for (int j = 0; j < RANK; ++j) b += Vh[(size_t)j * VOCAB + v];

#pragma unroll
  for (int t = 0; t < N_TOK; ++t)
    out[(size_t)t * VOCAB + v] = aSh[t] * b;
}

extern "C" void kernel_launch(void* const* d_in, const int* in_sizes, int n_in,
                              void* d_out, int out_size, void* d_ws, size_t ws_size,
                              hipStream_t stream) {
  const float* h  = (const float*)d_in[0];   // (32, 2048)
  const float* U  = (const float*)d_in[1];   // (2048, 64)
  const float* S  = (const float*)d_in[2];   // (64,)
  const float* Vh = (const float*)d_in[3];   // (64, 50257)
  float* out = (float*)d_out;                // (32, 50257)
  float* a   = (float*)d_ws;                 // 32 floats scratch

  fftlm_rowsum_wmma<<<1, 256, 0, stream>>>(h, U, S, a);

  const int blocks = (VOCAB + 255) / 256;
  fftlm_outer<<<blocks, 256, 0, stream>>>(Vh, a, out);
}